// EMAVectorQuantizer_65377992180178
// MI455X (gfx1250) — compile-verified
//
#include <hip/hip_runtime.h>
#include <math.h>

// Shapes fixed by the reference: B=32, D=128, T=2048, K=2048, N=B*T=65536.
#define BB 32
#define DD 128
#define TT 2048
#define KK 2048
#define NN (BB * TT)

typedef float v2f __attribute__((ext_vector_type(2)));
typedef float v8f __attribute__((ext_vector_type(8)));

// ---------------------------------------------------------------------------
// Kernel 0: init — zero segment sums / counters / scalars, compute ||e_k||^2.
// 256 blocks x 256 threads = 65536 threads; one wave per codebook row.
// ---------------------------------------------------------------------------
__global__ __launch_bounds__(256) void vq_init_kernel(
    const float* __restrict__ emb, float* __restrict__ enorm,
    float* __restrict__ sum_new, float* __restrict__ elem_new,
    float* __restrict__ scal)
{
    const int gtid = blockIdx.x * 256 + threadIdx.x;   // [0, 65536)
    // zero sum_new (K*D = 262144 floats) with float4 stores
    ((float4*)sum_new)[gtid] = make_float4(0.f, 0.f, 0.f, 0.f);

    const int wave = gtid >> 5;        // [0, 2048): one wave per code row
    const int lane = gtid & 31;
    const float* row = emb + (size_t)wave * DD;
    float s = 0.f;
#pragma unroll
    for (int q = 0; q < 4; ++q) {
        float v = row[lane + 32 * q];
        s += v * v;
    }
#pragma unroll
    for (int m = 16; m >= 1; m >>= 1) s += __shfl_xor(s, m, 32);
    if (lane == 0) {
        enorm[wave]    = s;
        elem_new[wave] = 0.f;
    }
    if (gtid == 0) { scal[0] = 0.f; scal[1] = 0.f; }
}

// ---------------------------------------------------------------------------
// Kernel 1: nearest-code search via V_WMMA_F32_16X16X4_F32.
// One wave owns TWO adjacent 16-row tiles of zf (32 rows): each B fragment
// (codebook) load feeds two WMMAs, and the B tile is split into two
// 16-fragment half-buffers (same 64 VGPRs as one full buffer) that are
// software-pipelined:
//     prefetch half1(kt) -> WMMA half0 -> prefetch half0(kt+1) -> WMMA half1
// so every load clause issues ~32 WMMAs before its first consumer and the
// L2 latency is hidden intra-wave (no WAR serialization between iterations).
// A-frags (2 x 16x128 fp32 = 128 VGPRs) stay register-resident for the whole
// K sweep; the 1 MB codebook is L2-resident (192 MB L2).
// dist(n,k) = ||e_k||^2 - 2 * <z_n, e_k>   (||z||^2 constant per row).
// Per-lane running (min, argmin); single cross-lane reduce at the end.
// ---------------------------------------------------------------------------
__global__ __launch_bounds__(256) void vq_argmin_kernel(
    const float* __restrict__ z, const float* __restrict__ emb,
    const float* __restrict__ enorm, int* __restrict__ idx_out)
{
    const int lane = threadIdx.x & 31;
    const int wid  = threadIdx.x >> 5;          // 8 waves per block
    const int m0   = (blockIdx.x * 8 + wid) * 32;   // 32 rows per wave
    const int b    = m0 >> 11;                  // T = 2048
    const int t0   = m0 & (TT - 1);
    const int hi   = lane >> 4;                 // half-wave: 0 or 1
    const int lm   = lane & 15;

    // A fragments: lane holds A[M = lm, K = {2*hi, 2*hi+1}] per 16x4 sub-tile.
    // zf[n, d] = z[b, d, t];  z layout [B, D, T].
    const float* zb = z + (size_t)b * (DD * TT) + t0 + lm;
    v2f a0[32], a1[32];
#pragma unroll
    for (int j = 0; j < 32; ++j) {
        const int d0 = 4 * j + 2 * hi;
        a0[j].x = zb[(size_t)d0 * TT];
        a0[j].y = zb[(size_t)(d0 + 1) * TT];
        a1[j].x = zb[(size_t)d0 * TT + 16];
        a1[j].y = zb[(size_t)(d0 + 1) * TT + 16];
    }

    float mv0[8], mv1[8];
    int   mi0[8], mi1[8];
#pragma unroll
    for (int r = 0; r < 8; ++r) {
        mv0[r] = 3.4e38f; mi0[r] = 0;
        mv1[r] = 3.4e38f; mi1[r] = 0;
    }

    // B fragment base for kt=0: lane holds B[K = {2*hi,2*hi+1}, N = lm]
    //   = embeddings[c, 4*j + 2*hi + {0,1}]  (contiguous b64 loads).
    const float* eb = emb + (size_t)lm * DD + 2 * hi;

    v2f bA[16], bB[16];                         // two half-tile buffers
#pragma unroll
    for (int j = 0; j < 16; ++j)                // preload half0 of kt=0
        bA[j] = *(const v2f*)(eb + 4 * j);

    for (int kt = 0; kt < KK / 16; ++kt) {
        const int c = kt * 16 + lm;             // this lane's code column
        v8f acc0 = {}, acc1 = {};

        // prefetch half1 of this kt (consumed after 32 WMMAs below)
#pragma unroll
        for (int j = 0; j < 16; ++j)
            bB[j] = *(const v2f*)(eb + 64 + 4 * j);

        // WMMA on half0
#pragma unroll
        for (int j = 0; j < 16; ++j) {
            acc0 = __builtin_amdgcn_wmma_f32_16x16x4_f32(
                false, a0[j], false, bA[j], (short)0, acc0, false, false);
            acc1 = __builtin_amdgcn_wmma_f32_16x16x4_f32(
                false, a1[j], false, bA[j], (short)0, acc1, false, false);
        }

        // prefetch half0 of the next kt (clamped on the last iteration so we
        // never read past the 1 MB codebook; the redundant reload is unused)
        const float* ebn = eb + ((kt < KK / 16 - 1) ? (size_t)(16 * DD) : 0);
#pragma unroll
        for (int j = 0; j < 16; ++j)
            bA[j] = *(const v2f*)(ebn + 4 * j);

        // WMMA on half1
#pragma unroll
        for (int j = 0; j < 16; ++j) {
            acc0 = __builtin_amdgcn_wmma_f32_16x16x4_f32(
                false, a0[16 + j], false, bB[j], (short)0, acc0, false, false);
            acc1 = __builtin_amdgcn_wmma_f32_16x16x4_f32(
                false, a1[16 + j], false, bB[j], (short)0, acc1, false, false);
        }

        const float en = enorm[c];
#pragma unroll
        for (int r = 0; r < 8; ++r) {
            // accX[r] = dot(z row M=r+8*hi (+16 for tile 1), code c)
            float d0 = en - 2.0f * acc0[r];
            if (d0 < mv0[r] || (d0 == mv0[r] && c < mi0[r])) { mv0[r] = d0; mi0[r] = c; }
            float d1 = en - 2.0f * acc1[r];
            if (d1 < mv1[r] || (d1 == mv1[r] && c < mi1[r])) { mv1[r] = d1; mi1[r] = c; }
        }
        eb = ebn;
    }

    // Cross-lane argmin within each 16-lane half (rows r+8*hi).
#pragma unroll
    for (int m = 8; m >= 1; m >>= 1) {
#pragma unroll
        for (int r = 0; r < 8; ++r) {
            float ov0 = __shfl_xor(mv0[r], m, 32);
            int   oi0 = __shfl_xor(mi0[r], m, 32);
            if (ov0 < mv0[r] || (ov0 == mv0[r] && oi0 < mi0[r])) { mv0[r] = ov0; mi0[r] = oi0; }
            float ov1 = __shfl_xor(mv1[r], m, 32);
            int   oi1 = __shfl_xor(mi1[r], m, 32);
            if (ov1 < mv1[r] || (ov1 == mv1[r] && oi1 < mi1[r])) { mv1[r] = ov1; mi1[r] = oi1; }
        }
    }
    if (lm == 0) {
#pragma unroll
        for (int r = 0; r < 8; ++r) {
            idx_out[m0 + 8 * hi + r]      = mi0[r];
            idx_out[m0 + 16 + 8 * hi + r] = mi1[r];
        }
    }
}

// ---------------------------------------------------------------------------
// Kernel 2: gather codewords -> straight-through output, enc loss,
// segment sums (native f32 atomics), counts.
// Block = 32 consecutive t (coalesced on the T axis) x 8 d-lanes.
// ---------------------------------------------------------------------------
__global__ __launch_bounds__(256) void vq_scatter_kernel(
    const float* __restrict__ z, const float* __restrict__ emb,
    const int* __restrict__ idx, float* __restrict__ zq_out,
    float* __restrict__ sum_new, float* __restrict__ elem_new,
    float* __restrict__ scal)
{
    const int tx = threadIdx.x & 31;        // t offset
    const int ty = threadIdx.x >> 5;        // d group
    const int n  = blockIdx.x * 32 + tx;
    const int b  = n >> 11;
    const int t  = n & (TT - 1);
    const int k  = idx[n];
    const size_t zbase = (size_t)b * (DD * TT) + t;

    float loss = 0.f;
#pragma unroll
    for (int j = 0; j < 16; ++j) {
        const int d  = ty + 8 * j;
        float zv = z[zbase + (size_t)d * TT];
        float ev = emb[k * DD + d];
        zq_out[zbase + (size_t)d * TT] = ev;   // forward value of straight-through
        float df = ev - zv;
        loss += df * df;
        __hip_atomic_fetch_add(&sum_new[k * DD + d], zv,
                               __ATOMIC_RELAXED, __HIP_MEMORY_SCOPE_AGENT);
    }
    if (ty == 0)
        __hip_atomic_fetch_add(&elem_new[k], 1.0f,
                               __ATOMIC_RELAXED, __HIP_MEMORY_SCOPE_AGENT);
#pragma unroll
    for (int m = 16; m >= 1; m >>= 1) loss += __shfl_xor(loss, m, 32);
    if (tx == 0)
        __hip_atomic_fetch_add(&scal[0], loss,
                               __ATOMIC_RELAXED, __HIP_MEMORY_SCOPE_AGENT);
}

// ---------------------------------------------------------------------------
// Kernel 3: EMA update, usage/dead-code replacement, dk partial sum.
// One thread per (k, d) element; K*D/256 blocks.
// ---------------------------------------------------------------------------
__global__ __launch_bounds__(256) void vq_ema_kernel(
    const float* __restrict__ emb, const float* __restrict__ emb_sum,
    const float* __restrict__ emb_elem, const float* __restrict__ emb_rand,
    const float* __restrict__ sum_new, const float* __restrict__ elem_new,
    float* __restrict__ out_emb_new, float* __restrict__ out_emb_sum,
    float* __restrict__ out_emb_elem, float* __restrict__ scal)
{
    const int g = blockIdx.x * 256 + threadIdx.x;   // [0, K*D)
    const int k = g >> 7;
    const int d = g & (DD - 1);

    float esum_n  = 0.99f * emb_sum[g]  + 0.01f * sum_new[g];
    float eelem_n = 0.99f * emb_elem[k] + 0.01f * elem_new[k];
    float usage   = (eelem_n >= 1.0f) ? 1.0f : 0.0f;
    float enew    = usage * (esum_n / eelem_n) + (1.0f - usage) * emb_rand[g];

    out_emb_new[g] = enew;
    out_emb_sum[g] = esum_n;
    if (d == 0) out_emb_elem[k] = eelem_n;

    float df = enew - emb[g];
    float sq = df * df;
#pragma unroll
    for (int m = 16; m >= 1; m >>= 1) sq += __shfl_xor(sq, m, 32);
    if ((threadIdx.x & 31) == 0)
        __hip_atomic_fetch_add(&scal[1], sq,
                               __ATOMIC_RELAXED, __HIP_MEMORY_SCOPE_AGENT);
}

// ---------------------------------------------------------------------------
// Kernel 4: finalize scalars — entropy of code usage, dk, enc loss.
// ---------------------------------------------------------------------------
__global__ __launch_bounds__(256) void vq_finalize_kernel(
    const float* __restrict__ elem_new, const float* __restrict__ scal,
    float* __restrict__ out_loss, float* __restrict__ out_entropy,
    float* __restrict__ out_dk)
{
    __shared__ float red[256];
    const int tid = threadIdx.x;

    float s = 0.f;
    for (int i = tid; i < KK; i += 256) s += elem_new[i];
    red[tid] = s;
    __syncthreads();
    for (int w = 128; w >= 1; w >>= 1) {
        if (tid < w) red[tid] += red[tid + w];
        __syncthreads();
    }
    const float total = red[0];
    __syncthreads();

    float h = 0.f;
    for (int i = tid; i < KK; i += 256) {
        float p = elem_new[i] / total;
        h += p * logf(p + 1e-8f);
    }
    red[tid] = h;
    __syncthreads();
    for (int w = 128; w >= 1; w >>= 1) {
        if (tid < w) red[tid] += red[tid + w];
        __syncthreads();
    }
    if (tid == 0) {
        out_entropy[0] = expf(-red[0]);
        out_dk[0]      = sqrtf(scal[1] / (float)(KK * DD));
        out_loss[0]    = scal[0];
    }
}

// ---------------------------------------------------------------------------
extern "C" void kernel_launch(void* const* d_in, const int* in_sizes, int n_in,
                              void* d_out, int out_size, void* d_ws, size_t ws_size,
                              hipStream_t stream) {
    const float* z          = (const float*)d_in[0];   // [B, D, T]
    const float* embeddings = (const float*)d_in[1];   // [K, D]
    const float* emb_sum    = (const float*)d_in[2];   // [K, D]
    const float* emb_elem   = (const float*)d_in[3];   // [K]
    const float* emb_rand   = (const float*)d_in[4];   // [K, D]

    float* out        = (float*)d_out;
    float* o_zq       = out;                                   // B*D*T
    float* o_loss     = out + (size_t)BB * DD * TT;            // 1
    float* o_emb_new  = o_loss + 1;                            // K*D
    float* o_emb_sum  = o_emb_new + (size_t)KK * DD;           // K*D
    float* o_emb_elem = o_emb_sum + (size_t)KK * DD;           // K
    float* o_entropy  = o_emb_elem + KK;                       // 1
    float* o_dk       = o_entropy + 1;                         // 1

    char*  ws       = (char*)d_ws;
    int*   idx      = (int*)ws;                                // N ints
    float* enorm    = (float*)(ws + (size_t)NN * 4);           // K floats
    float* sum_new  = (float*)(ws + (size_t)NN * 4 + KK * 4);  // K*D floats
    float* elem_new = (float*)(ws + (size_t)NN * 4 + KK * 4 + (size_t)KK * DD * 4);
    float* scal     = elem_new + KK;                           // [loss, dk_sq]

    vq_init_kernel<<<256, 256, 0, stream>>>(embeddings, enorm, sum_new, elem_new, scal);
    vq_argmin_kernel<<<NN / 32 / 8, 256, 0, stream>>>(z, embeddings, enorm, idx);
    vq_scatter_kernel<<<NN / 32, 256, 0, stream>>>(z, embeddings, idx, o_zq,
                                                   sum_new, elem_new, scal);
    vq_ema_kernel<<<KK * DD / 256, 256, 0, stream>>>(embeddings, emb_sum, emb_elem,
                                                     emb_rand, sum_new, elem_new,
                                                     o_emb_new, o_emb_sum, o_emb_elem, scal);
    vq_finalize_kernel<<<1, 256, 0, stream>>>(elem_new, scal, o_loss, o_entropy, o_dk);
}